// DirectedGATLayerInversed_36009005809889
// MI455X (gfx1250) — compile-verified
//
#include <hip/hip_runtime.h>
#include <math.h>

#define GAT_F 64
#define GAT_ALPHA 0.2f

typedef __attribute__((ext_vector_type(2))) float v2f;
typedef __attribute__((ext_vector_type(8))) float v8f;

// ---------- order-preserving float <-> uint encoding for atomic max ----------
__device__ __forceinline__ unsigned enc_f32(float f) {
    unsigned u = __float_as_uint(f);
    return (u & 0x80000000u) ? ~u : (u | 0x80000000u);
}
__device__ __forceinline__ float dec_f32(unsigned u) {
    unsigned b = (u & 0x80000000u) ? (u ^ 0x80000000u) : ~u;
    return __uint_as_float(b);
}

// ---------- kernel 0: zero output, init segment-max / denom -----------------
__global__ void gat_init(float* __restrict__ out, unsigned* __restrict__ m_u,
                         float* __restrict__ denom, int NF, int N) {
    int i = blockIdx.x * blockDim.x + threadIdx.x;
    if (i < NF) out[i] = 0.0f;
    if (i < N) {
        m_u[i]   = 0x007FFFFFu;   // enc_f32(-inf)
        denom[i] = 0.0f;
    }
}

// ---------- kernel 1: Wh = h @ W^T + Wb  (WMMA f32 16x16x4), fused s1/s2 ----
// One wave handles 16 nodes x all 64 output features (4 accumulator tiles).
__global__ __launch_bounds__(256) void gat_gemm(
    const float* __restrict__ h, const float* __restrict__ W,
    const float* __restrict__ Wb, const float* __restrict__ a,
    float* __restrict__ Wh, float* __restrict__ s1, float* __restrict__ s2,
    int N) {
    const int lane   = threadIdx.x & 31;
    const int waveId = (blockIdx.x * blockDim.x + threadIdx.x) >> 5;
    const int ntiles = N >> 4;                 // N divisible by 16 (100000)
    if (waveId >= ntiles) return;              // wave-uniform exit: EXEC stays all-1s

    const int row0  = waveId << 4;
    const int laneM = lane & 15;               // row within A / col within B
    const int laneH = lane >> 4;               // K-half selector

    v8f c0 = {}, c1 = {}, c2 = {}, c3 = {};

    const float* hrow = h + (size_t)(row0 + laneM) * GAT_F + 2 * laneH;
    const float* Wr0  = W + (size_t)(laneM +  0) * GAT_F + 2 * laneH;
    const float* Wr1  = W + (size_t)(laneM + 16) * GAT_F + 2 * laneH;
    const float* Wr2  = W + (size_t)(laneM + 32) * GAT_F + 2 * laneH;
    const float* Wr3  = W + (size_t)(laneM + 48) * GAT_F + 2 * laneH;

#pragma unroll
    for (int kk = 0; kk < 16; ++kk) {
        const int k = kk * 4;
        v2f A;  A.x = hrow[k]; A.y = hrow[k + 1];          // A: row M=laneM, K pair
        v2f B0; B0.x = Wr0[k]; B0.y = Wr0[k + 1];          // B: W^T[k][n] = W[n][k]
        v2f B1; B1.x = Wr1[k]; B1.y = Wr1[k + 1];
        v2f B2; B2.x = Wr2[k]; B2.y = Wr2[k + 1];
        v2f B3; B3.x = Wr3[k]; B3.y = Wr3[k + 1];
        c0 = __builtin_amdgcn_wmma_f32_16x16x4_f32(false, A, false, B0, (short)0, c0, false, false);
        c1 = __builtin_amdgcn_wmma_f32_16x16x4_f32(false, A, false, B1, (short)0, c1, false, false);
        c2 = __builtin_amdgcn_wmma_f32_16x16x4_f32(false, A, false, B2, (short)0, c2, false, false);
        c3 = __builtin_amdgcn_wmma_f32_16x16x4_f32(false, A, false, B3, (short)0, c3, false, false);
    }

    // Add bias (one output column per lane per tile)
    const float b0 = Wb[laneM], b1 = Wb[laneM + 16], b2 = Wb[laneM + 32], b3 = Wb[laneM + 48];
#pragma unroll
    for (int r = 0; r < 8; ++r) { c0[r] += b0; c1[r] += b1; c2[r] += b2; c3[r] += b3; }

    // Store Wh: C layout -> VGPR r holds row M=r (lanes 0-15) / M=r+8 (lanes 16-31)
#pragma unroll
    for (int r = 0; r < 8; ++r) {
        const size_t row = (size_t)(row0 + r + 8 * laneH);
        Wh[row * GAT_F + laneM +  0] = c0[r];
        Wh[row * GAT_F + laneM + 16] = c1[r];
        Wh[row * GAT_F + laneM + 32] = c2[r];
        Wh[row * GAT_F + laneM + 48] = c3[r];
    }

    // Fused per-node attention scalars: s1 = Wh . a[:F], s2 = Wh . a[F:]
    const float as0 = a[laneM], as1 = a[laneM + 16], as2 = a[laneM + 32], as3 = a[laneM + 48];
    const float ad0 = a[GAT_F + laneM],      ad1 = a[GAT_F + laneM + 16],
                ad2 = a[GAT_F + laneM + 32], ad3 = a[GAT_F + laneM + 48];
    float p1[8], p2[8];
#pragma unroll
    for (int r = 0; r < 8; ++r) {
        p1[r] = c0[r] * as0 + c1[r] * as1 + c2[r] * as2 + c3[r] * as3;
        p2[r] = c0[r] * ad0 + c1[r] * ad1 + c2[r] * ad2 + c3[r] * ad3;
    }
    // Reduce across the 16 lanes of each half (xor masks 1,2,4,8 stay in-half)
#pragma unroll
    for (int off = 1; off < 16; off <<= 1) {
#pragma unroll
        for (int r = 0; r < 8; ++r) {
            p1[r] += __shfl_xor(p1[r], off, 32);
            p2[r] += __shfl_xor(p2[r], off, 32);
        }
    }
    if (laneM == 0) {
#pragma unroll
        for (int r = 0; r < 8; ++r) {
            const int row = row0 + r + 8 * laneH;
            s1[row] = p1[r];
            s2[row] = p2[r];
        }
    }
}

// ---------- kernel 2: per-edge logits + segment max -------------------------
__global__ void gat_edge_e(const int* __restrict__ src, const int* __restrict__ dst,
                           const float* __restrict__ s1, const float* __restrict__ s2,
                           const float* __restrict__ ab, float* __restrict__ ebuf,
                           unsigned* __restrict__ m_u, int E) {
    int i = blockIdx.x * blockDim.x + threadIdx.x;
    if (i >= E) return;
    const int s = src[i], d = dst[i];
    float e = s1[s] + s2[d] + ab[0];
    e = (e >= 0.0f) ? e : GAT_ALPHA * e;      // leaky_relu
    ebuf[i] = e;
    atomicMax(&m_u[d], enc_f32(e));
}

// ---------- kernel 3: exp(e - m[dst]) + segment sum -------------------------
__global__ void gat_edge_ex(const int* __restrict__ dst, const unsigned* __restrict__ m_u,
                            float* __restrict__ ebuf, float* __restrict__ denom, int E) {
    int i = blockIdx.x * blockDim.x + threadIdx.x;
    if (i >= E) return;
    const int d = dst[i];
    const float ex = __expf(ebuf[i] - dec_f32(m_u[d]));
    ebuf[i] = ex;
    atomicAdd(&denom[d], ex);
}

// ---------- kernel 4: h'[dst] += attn * Wh[src]  (wave per edge) ------------
__global__ __launch_bounds__(256) void gat_scatter(
    const int* __restrict__ src, const int* __restrict__ dst,
    const float* __restrict__ ebuf, const float* __restrict__ denom,
    const float* __restrict__ Wh, float* __restrict__ out, int E) {
    const long long tid = (long long)blockIdx.x * blockDim.x + threadIdx.x;
    const int edge = (int)(tid >> 5);
    if (edge >= E) return;
    const int lane = (int)(tid & 31);
    const int s = src[edge], d = dst[edge];
    const float dn = denom[d];
    const float attn = ebuf[edge] / ((dn > 0.0f) ? dn : 1.0f);
    const float2 w = ((const float2*)(Wh + (size_t)s * GAT_F))[lane];
    float* o = out + (size_t)d * GAT_F + 2 * lane;
    atomicAdd(o,     attn * w.x);
    atomicAdd(o + 1, attn * w.y);
}

extern "C" void kernel_launch(void* const* d_in, const int* in_sizes, int n_in,
                              void* d_out, int out_size, void* d_ws, size_t ws_size,
                              hipStream_t stream) {
    const float* h  = (const float*)d_in[0];
    const float* W  = (const float*)d_in[1];
    const float* Wb = (const float*)d_in[2];
    const float* a  = (const float*)d_in[3];
    const float* ab = (const float*)d_in[4];
    const int* src  = (const int*)d_in[5];
    const int* dst  = (const int*)d_in[6];

    const int N = in_sizes[0] / GAT_F;   // 100000
    const int E = in_sizes[5];           // 1000000
    const int NF = N * GAT_F;

    float* out = (float*)d_out;

    // workspace layout
    float*    Wh    = (float*)d_ws;                 // N*F
    float*    s1    = Wh + (size_t)NF;              // N
    float*    s2    = s1 + N;                       // N
    unsigned* m_u   = (unsigned*)(s2 + N);          // N
    float*    denom = (float*)(m_u + N);            // N
    float*    ebuf  = denom + N;                    // E

    // 0: init
    gat_init<<<(NF + 255) / 256, 256, 0, stream>>>(out, m_u, denom, NF, N);

    // 1: WMMA GEMM + fused attention scalars (one wave per 16 nodes)
    {
        const int waves  = (N + 15) / 16;
        const int blocks = (waves + 7) / 8;          // 8 waves (256 threads) per block
        gat_gemm<<<blocks, 256, 0, stream>>>(h, W, Wb, a, Wh, s1, s2, N);
    }

    // 2: edge logits + segment max
    gat_edge_e<<<(E + 255) / 256, 256, 0, stream>>>(src, dst, s1, s2, ab, ebuf, m_u, E);

    // 3: exp + segment sum
    gat_edge_ex<<<(E + 255) / 256, 256, 0, stream>>>(dst, m_u, ebuf, denom, E);

    // 4: weighted scatter (one wave per edge)
    {
        const long long threads = (long long)E * 32;
        const int blocks = (int)((threads + 255) / 256);
        gat_scatter<<<blocks, 256, 0, stream>>>(src, dst, ebuf, denom, Wh, out, E);
    }
}